// KMeans_16947940950566
// MI455X (gfx1250) — compile-verified
//
#include <hip/hip_runtime.h>
#include <stdint.h>

#define B_ 4
#define I_ 4
#define BI 16
#define N_ 4096
#define K_ 256
#define D_ 128
#define KD (K_*D_)
#define BIKD (BI*KD)
#define MAX_ITER_ 10
#define TOL_ 1e-4f

// LDS tile: 16 cluster rows x 128 dims, padded to 130-float row stride so the
// 16 lanes reading row*stride + d hit distinct banks (TDM pad feature writes
// the same layout: 2 dwords of pad after every 128 dwords).
#define TILE_ROWS 16
#define LDS_STRIDE 130
#define TILE_FLOATS (TILE_ROWS * LDS_STRIDE)

typedef __attribute__((ext_vector_type(2))) float v2f;
typedef __attribute__((ext_vector_type(8))) float v8f;
typedef __attribute__((ext_vector_type(4))) unsigned int u32x4;
typedef __attribute__((ext_vector_type(8))) int i32x8;
typedef __attribute__((ext_vector_type(4))) int i32x4;

#if __has_builtin(__builtin_amdgcn_tensor_load_to_lds)
#define HAVE_TDM 1
#else
#define HAVE_TDM 0
#endif

#if HAVE_TDM
// Issue a TDM load of one 16x128 fp32 tile (row stride 128 in memory) into LDS
// at lds_byte_addr, with 2-dword padding per 128-dword row (LDS stride 130).
__device__ __forceinline__ void tdm_load_tile(const float* gsrc,
                                              unsigned lds_byte_addr) {
  unsigned long long ga = (unsigned long long)(uintptr_t)gsrc;
  u32x4 g0;
  g0[0] = 1u;                                      // count=1, user descriptor
  g0[1] = lds_byte_addr;                           // lds_addr
  g0[2] = (unsigned)(ga & 0xFFFFFFFFu);            // global_addr[31:0]
  g0[3] = (unsigned)((ga >> 32) & 0x01FFFFFFu)     // global_addr[56:32]
          | (2u << 30);                            // type=2 ("image")
  i32x8 g1;
  g1[0] = (int)((2u << 16)       // data_size = 4 bytes
                | (1u << 20)     // pad_enable
                | (6u << 22)     // pad_interval: 128 dwords
                | (1u << 25));   // pad_amount: 2 dwords
  g1[1] = (int)(128u << 16);     // tensor_dim0 = 128 (bits 79:48)
  g1[2] = (int)(16u << 16);      // tensor_dim1 = 16  (bits 111:80)
  g1[3] = (int)(128u << 16);     // tile_dim0 = 128   (bits 127:112)
  g1[4] = 16;                    // tile_dim1 = 16, tile_dim2 = 0
  g1[5] = 128;                   // tensor_dim0_stride = 128 (bits 207:160)
  g1[6] = 0;                     // stride hi / tensor_dim1_stride lo (2D: 0)
  g1[7] = 0;
  i32x4 z4 = {0, 0, 0, 0};
#if defined(__clang_major__) && __clang_major__ >= 23
  i32x8 z8 = {0, 0, 0, 0, 0, 0, 0, 0};
  __builtin_amdgcn_tensor_load_to_lds(g0, g1, z4, z4, z8, 0);
#else
  __builtin_amdgcn_tensor_load_to_lds(g0, g1, z4, z4, 0);
#endif
}

__device__ __forceinline__ void tdm_wait() {
#if __has_builtin(__builtin_amdgcn_s_wait_tensorcnt)
  __builtin_amdgcn_s_wait_tensorcnt(0);
#else
  asm volatile("s_wait_tensorcnt 0x0" ::: "memory");
#endif
}
#endif  // HAVE_TDM

// ---------------------------------------------------------------------------
// init: centers_cur = centers_in; done = 0; inertia = 0
// ---------------------------------------------------------------------------
__global__ __launch_bounds__(256)
void init_kernel(const float* __restrict__ c_in, float* __restrict__ centers,
                 int* __restrict__ done, float* __restrict__ inertia) {
  int gid = blockIdx.x * blockDim.x + threadIdx.x;
  if (gid < BIKD) centers[gid] = c_in[gid];
  if (gid < BI) { done[gid] = 0; inertia[gid] = 0.f; }
}

// ---------------------------------------------------------------------------
// prep: zero sums/counts, compute c2[bi,k] = sum_d centers[bi,k,d]^2
// ---------------------------------------------------------------------------
__global__ __launch_bounds__(256)
void prep_kernel(const float* __restrict__ centers, float* __restrict__ sums,
                 float* __restrict__ counts, float* __restrict__ c2) {
  int gid = blockIdx.x * blockDim.x + threadIdx.x;
  int stride = gridDim.x * blockDim.x;
  for (int e = gid; e < BIKD; e += stride) sums[e] = 0.f;
  if (gid < BI * K_) {
    counts[gid] = 0.f;
    const float4* row = (const float4*)(centers + (size_t)gid * D_);
    float s = 0.f;
#pragma unroll
    for (int q = 0; q < D_ / 4; ++q) {
      float4 v = row[q];
      s += v.x * v.x + v.y * v.y + v.z * v.z + v.w * v.w;
    }
    c2[gid] = s;
  }
}

// ---------------------------------------------------------------------------
// assign: fp32 WMMA cross-term GEMM with TDM-staged, double-buffered centers
// tiles in LDS. Each wave: 16 points (B operand in registers), loops 16
// cluster tiles shared by all 8 waves of the block via LDS.
// MODE 0: Lloyd E-step + atomic scatter of centroid partial sums
// MODE 1: final pass -> labels + inertia
// ---------------------------------------------------------------------------
template <int MODE>
__global__ __launch_bounds__(256)
void assign_kernel(const float* __restrict__ x, const float* __restrict__ centers,
                   const float* __restrict__ c2, float* __restrict__ sums,
                   float* __restrict__ counts, const int* __restrict__ done,
                   int* __restrict__ labels, float* __restrict__ inertia) {
  const int bi = blockIdx.y;
  if (MODE == 0 && done[bi]) return;
  const int b = bi >> 2;                 // I_ == 4
  const int lane = threadIdx.x & 31;
  const int wave = threadIdx.x >> 5;
  const int lo = lane & 15;              // point column / cluster row in tile
  const int hi = lane >> 4;              // half-wave selector
  const int kb = hi << 1;                // K-pair base within a K=4 chunk
  const int n = blockIdx.x * 128 + wave * 16 + lo;

  const float* xb = x + (size_t)b * N_ * D_;
  const float* C = centers + (size_t)bi * KD;
  const float* c2p = c2 + bi * K_;

  __shared__ float smem[2][TILE_FLOATS];

#if HAVE_TDM
  // prologue: DMA tile 0 into buffer 0 while we load the point tile
  if (wave == 0) tdm_load_tile(C, (unsigned)(uintptr_t)&smem[0][0]);
#endif

  // B operand (point tile) pinned in registers, reused by all 16 cluster tiles
  v2f breg[32];
  {
    const float* xr = xb + (size_t)n * D_ + kb;
#pragma unroll
    for (int c = 0; c < 32; ++c) breg[c] = *(const v2f*)(xr + 4 * c);
  }

  float bestv = 3.0e38f;
  int bestk = 0;

  for (int t = 0; t < 16; ++t) {
    const int cur = t & 1;
#if HAVE_TDM
    if (wave == 0) tdm_wait();           // tile t landed in smem[cur]
    __syncthreads();                     // visible to all waves
    if (t < 15 && wave == 0)             // overlap tile t+1 DMA with compute
      tdm_load_tile(C + (size_t)(t + 1) * TILE_ROWS * D_,
                    (unsigned)(uintptr_t)&smem[(t + 1) & 1][0]);
#else
    // cooperative staged copy fallback (same padded layout)
    for (int q = threadIdx.x; q < TILE_ROWS * (D_ / 2); q += 256) {
      int row = q >> 6;                  // 64 float2 per row
      int col = (q & 63) * 2;
      *(v2f*)&smem[cur][row * LDS_STRIDE + col] =
          *(const v2f*)(C + (size_t)(t * TILE_ROWS + row) * D_ + col);
    }
    __syncthreads();
#endif

    v8f acc = {0.f, 0.f, 0.f, 0.f, 0.f, 0.f, 0.f, 0.f};
    const float* ar = &smem[cur][lo * LDS_STRIDE + kb];
#pragma unroll
    for (int c = 0; c < 32; ++c) {
      v2f a = *(const v2f*)(ar + 4 * c);
      acc = __builtin_amdgcn_wmma_f32_16x16x4_f32(false, a, false, breg[c],
                                                  (short)0, acc, false, false);
    }
    // lane holds 8 cluster rows (m = mbase..mbase+7) for point column n
    const int mbase = t * 16 + hi * 8;
    const float4 ca = *(const float4*)(c2p + mbase);
    const float4 cb = *(const float4*)(c2p + mbase + 4);
    float cv[8] = {ca.x, ca.y, ca.z, ca.w, cb.x, cb.y, cb.z, cb.w};
#pragma unroll
    for (int j = 0; j < 8; ++j) {
      float d2 = cv[j] - 2.0f * acc[j];  // ||x||^2 constant per point: omit
      int m = mbase + j;
      if (d2 < bestv) { bestv = d2; bestk = m; }
    }
  }

  // combine with partner lane (lane ^ 16) holding the other 8 rows per tile
  {
    float pv = __shfl_xor(bestv, 16, 32);
    int pk = __shfl_xor(bestk, 16, 32);
    if (pv < bestv || (pv == bestv && pk < bestk)) { bestv = pv; bestk = pk; }
  }

  if (MODE == 0) {
    // scatter centroid partial sums: 32 lanes cover D=128 with 4 dims each
    for (int p = 0; p < 16; ++p) {
      int k = __shfl(bestk, p, 32);
      int np = blockIdx.x * 128 + wave * 16 + p;
      const float4 v = *(const float4*)(xb + (size_t)np * D_ + lane * 4);
      float* sr = sums + ((size_t)bi * K_ + k) * D_ + lane * 4;
      atomicAdd(sr + 0, v.x);
      atomicAdd(sr + 1, v.y);
      atomicAdd(sr + 2, v.z);
      atomicAdd(sr + 3, v.w);
      if (lane == 0) atomicAdd(&counts[bi * K_ + k], 1.0f);
    }
  } else {
    // final: add back ||x||^2 (each half-wave holds half the dims of a point)
    float own = 0.f;
#pragma unroll
    for (int c = 0; c < 32; ++c)
      own += breg[c].x * breg[c].x + breg[c].y * breg[c].y;
    float x2 = own + __shfl_xor(own, 16, 32);
    float mind = bestv + x2;
    if (hi == 0) {
      labels[bi * N_ + n] = bestk;
      atomicAdd(&inertia[bi], mind);
    }
  }
}

// ---------------------------------------------------------------------------
// update: new centers, Frobenius shift, convergence freeze (per (b,i))
// ---------------------------------------------------------------------------
__global__ __launch_bounds__(256)
void update_kernel(float* __restrict__ centers, const float* __restrict__ sums,
                   const float* __restrict__ counts, int* __restrict__ done) {
  const int bi = blockIdx.x;
  if (done[bi]) return;
  __shared__ float red[256];
  float local = 0.f;
  for (int e = threadIdx.x; e < KD; e += 256) {
    int k = e >> 7;  // D_ == 128
    float cnt = counts[bi * K_ + k];
    size_t idx = (size_t)bi * KD + e;
    float oldv = centers[idx];
    float nv = (cnt > 0.f) ? sums[idx] / fmaxf(cnt, 1.f) : oldv;
    centers[idx] = nv;
    float df = nv - oldv;
    local += df * df;
  }
  red[threadIdx.x] = local;
  __syncthreads();
  for (int s = 128; s > 0; s >>= 1) {
    if (threadIdx.x < s) red[threadIdx.x] += red[threadIdx.x + s];
    __syncthreads();
  }
  if (threadIdx.x == 0 && red[0] < TOL_) done[bi] = 1;
}

// ---------------------------------------------------------------------------
// output: per-batch best restart -> labels(float), centers, inertia
// ---------------------------------------------------------------------------
__global__ __launch_bounds__(256)
void output_kernel(const int* __restrict__ labels, const float* __restrict__ centers,
                   const float* __restrict__ inertia, float* __restrict__ out) {
  const int b = blockIdx.x;
  float bv = 3.0e38f;
  int best = 0;
  for (int i = 0; i < I_; ++i) {
    float v = inertia[b * I_ + i];
    if (v < bv) { bv = v; best = i; }
  }
  const int bi = b * I_ + best;
  for (int nn = threadIdx.x; nn < N_; nn += 256)
    out[b * N_ + nn] = (float)labels[bi * N_ + nn];
  float* oc = out + B_ * N_;
  for (int e = threadIdx.x; e < KD; e += 256)
    oc[(size_t)b * KD + e] = centers[(size_t)bi * KD + e];
  if (threadIdx.x == 0) out[B_ * N_ + B_ * KD + b] = bv;
}

// ---------------------------------------------------------------------------
extern "C" void kernel_launch(void* const* d_in, const int* in_sizes, int n_in,
                              void* d_out, int out_size, void* d_ws, size_t ws_size,
                              hipStream_t stream) {
  const float* x = (const float*)d_in[0];
  const float* c_in = (const float*)d_in[1];

  float* ws = (float*)d_ws;
  float* centers = ws;                    // BIKD
  float* sums    = centers + BIKD;        // BIKD
  float* counts  = sums + BIKD;           // BI*K
  float* c2      = counts + BI * K_;      // BI*K
  float* inertia = c2 + BI * K_;          // BI
  int*   done    = (int*)(inertia + BI);  // BI
  int*   labels  = done + BI;             // BI*N
  float* out = (float*)d_out;

  init_kernel<<<dim3((BIKD + 255) / 256), 256, 0, stream>>>(c_in, centers, done,
                                                            inertia);
  for (int it = 0; it < MAX_ITER_; ++it) {
    prep_kernel<<<256, 256, 0, stream>>>(centers, sums, counts, c2);
    assign_kernel<0><<<dim3(N_ / 128, BI), 256, 0, stream>>>(
        x, centers, c2, sums, counts, done, labels, inertia);
    update_kernel<<<BI, 256, 0, stream>>>(centers, sums, counts, done);
  }

  prep_kernel<<<256, 256, 0, stream>>>(centers, sums, counts, c2);
  assign_kernel<1><<<dim3(N_ / 128, BI), 256, 0, stream>>>(
      x, centers, c2, sums, counts, done, labels, inertia);
  output_kernel<<<B_, 256, 0, stream>>>(labels, centers, inertia, out);
}